// TensorProductLayer_13898514170591
// MI455X (gfx1250) — compile-verified
//
#include <hip/hip_runtime.h>

typedef float v2f __attribute__((ext_vector_type(2)));
typedef float v4f __attribute__((ext_vector_type(4)));
typedef float v8f __attribute__((ext_vector_type(8)));

// ======================================================================
// Compile-time Clebsch-Gordan coefficients (Racah formula), L_MAX = 3.
// ======================================================================
constexpr double FACT[14] = {1.,1.,2.,6.,24.,120.,720.,5040.,40320.,362880.,
                             3628800.,39916800.,479001600.,6227020800.};

constexpr int iabs_(int a) { return a < 0 ? -a : a; }

constexpr double csqrt_(double x) {
  if (x <= 0.) return 0.;
  double g = x, r = 1.;
  while (g > 4.)   { g *= 0.25; r *= 2.; }
  while (g < 0.25) { g *= 4.;   r *= 0.5; }
  double y = g;
  for (int i = 0; i < 16; ++i) y = 0.5 * (y + g / y);
  return r * y;
}

constexpr double cg_(int l1, int l2, int l3, int m1, int m2, int m3) {
  if (m1 + m2 != m3) return 0.;
  if (l3 < iabs_(l1 - l2) || l3 > l1 + l2) return 0.;
  if (iabs_(m1) > l1 || iabs_(m2) > l2 || iabs_(m3) > l3) return 0.;
  double p2 = (2 * l3 + 1) * FACT[l3 + l1 - l2] * FACT[l3 - l1 + l2] *
              FACT[l1 + l2 - l3] / FACT[l1 + l2 + l3 + 1] *
              FACT[l3 + m3] * FACT[l3 - m3] * FACT[l1 - m1] *
              FACT[l1 + m1] * FACT[l2 - m2] * FACT[l2 + m2];
  double pref = csqrt_(p2);
  double s = 0.;
  for (int k = 0; k <= l1 + l2 - l3; ++k) {
    int a0 = k, a1 = l1 + l2 - l3 - k, a2 = l1 - m1 - k, a3 = l2 + m2 - k,
        a4 = l3 - l2 + m1 + k, a5 = l3 - l1 - m2 + k;
    if (a0 < 0 || a1 < 0 || a2 < 0 || a3 < 0 || a4 < 0 || a5 < 0) continue;
    double d = FACT[a0] * FACT[a1] * FACT[a2] * FACT[a3] * FACT[a4] * FACT[a5];
    s += ((k & 1) ? -1. : 1.) / d;
  }
  return pref * s;
}

// Global irrep index -> (l, m).  Offsets: l0->[0], l1->[1,3], l2->[4,8], l3->[9,15]
struct LM { int l; int m; };
constexpr LM dec_(int g) {
  if (g == 0) return {0, 0};
  if (g < 4)  return {1, g - 2};
  if (g < 9)  return {2, g - 6};
  return {3, g - 12};
}

// W[(I*16+J)*16 + K] : fused CG tensor so out[p,K] = sum_{I,J} f1[p,I] f2[p,J] W[I,J,K]
struct alignas(16) WTab { float w[256 * 16]; };
constexpr WTab make_wtab() {
  WTab t{};
  for (int I = 0; I < 16; ++I)
    for (int J = 0; J < 16; ++J)
      for (int K = 0; K < 16; ++K) {
        LM a = dec_(I), b = dec_(J), c = dec_(K);
        double v = cg_(a.l, b.l, c.l, a.m, b.m, c.m);
        if (v < 1e-10 && v > -1e-10) v = 0.;   // match reference thresholding
        t.w[(I * 16 + J) * 16 + K] = (float)v;
      }
  return t;
}

__constant__ WTab g_W = make_wtab();

// ======================================================================
// One wave32 handles 16 points via 64x V_WMMA_F32_16X16X4_F32.
//   A tile  : P[16 points x 4 pairs]  (pair products, 2 v_mul/step)
//   B tile  : W[4 pairs x 16 outputs] (constant, staged in LDS, 1 ds/step)
//   D tile  : out[16 points x 16 outputs] -> coalesced b32 stores
// VGPR layouts per CDNA5 ISA 7.12.2 (A: lane halves hold K=0/1 vs K=2/3;
// D: VGPR v holds rows v / v+8; B mirrors A).
// Ragged tail handled by clamping the last tile base to NC-16: the overlap
// is recomputed identically by two waves, so duplicate stores are benign and
// EXEC stays all-ones for every WMMA (clamp is wave-uniform).
// ======================================================================
__global__ __launch_bounds__(256) void cg_tp_wmma(
    const float* __restrict__ f10, const float* __restrict__ f11,
    const float* __restrict__ f12, const float* __restrict__ f13,
    const float* __restrict__ f20, const float* __restrict__ f21,
    const float* __restrict__ f22, const float* __restrict__ f23,
    float* __restrict__ out, int ntiles, int NC)
{
  __shared__ alignas(16) float sW[256 * 16];
  // Vectorized table stage-in: 4x b128 load + 4x ds_store_b128 per thread.
  {
    const v4f* __restrict__ src = (const v4f*)g_W.w;
    v4f* dst = (v4f*)sW;
    for (int i = threadIdx.x; i < 1024; i += 256) dst[i] = src[i];
  }
  __syncthreads();

  const int lane = threadIdx.x & 31;
  const int tile = blockIdx.x * 8 + (threadIdx.x >> 5);
  if (tile >= ntiles) return;          // wave-uniform: EXEC stays all-ones

  int base = tile * 16;                // wave-uniform clamp for ragged tail
  if (base + 16 > NC) base = NC - 16;

  const int  col = lane & 15;
  const bool lo  = lane < 16;          // lanes 0-15: K=0/1, lanes 16-31: K=2/3
  const int  r   = base + col;         // this lane's point (row)

  // Gather this row's full 16-component irrep vectors (coalesced across lanes).
  float F1[16], F2[16];
  F1[0] = f10[r];  F2[0] = f20[r];
#pragma unroll
  for (int m = 0; m < 3; ++m) { F1[1 + m] = f11[r * 3 + m]; F2[1 + m] = f21[r * 3 + m]; }
#pragma unroll
  for (int m = 0; m < 5; ++m) { F1[4 + m] = f12[r * 5 + m]; F2[4 + m] = f22[r * 5 + m]; }
#pragma unroll
  for (int m = 0; m < 7; ++m) { F1[9 + m] = f13[r * 7 + m]; F2[9 + m] = f23[r * 7 + m]; }

  // Hoisted lane-half selection of f2 operands: only 8 distinct values are
  // ever needed (J0 cycles over {0,4,8,12}).  8 cndmasks total, none in loop.
  float S[8];
#pragma unroll
  for (int g = 0; g < 4; ++g) {
    S[2 * g]     = lo ? F2[4 * g]     : F2[4 * g + 2];
    S[2 * g + 1] = lo ? F2[4 * g + 1] : F2[4 * g + 3];
  }

  // B-operand LDS base for this lane: row (4s + off2 + {0,1}), col = output k
  const int bbase = (lo ? 0 : 32) + col;   // off2*16 + col, off2 in {0,2}

  v8f acc0 = {}; v8f acc1 = {};            // two chains for ILP
#pragma unroll
  for (int s = 0; s < 64; ++s) {
    const int I = s >> 2;                  // pairs 4s..4s+3 share f1 index I
    const int g = s & 3;                   // J0 = 4*g
    v2f a, b;
    a.x = F1[I] * S[2 * g];
    a.y = F1[I] * S[2 * g + 1];
    b.x = sW[s * 64 + bbase];
    b.y = sW[s * 64 + 16 + bbase];
    if (s & 1)
      acc1 = __builtin_amdgcn_wmma_f32_16x16x4_f32(false, a, false, b,
                                                   (short)0, acc1, false, false);
    else
      acc0 = __builtin_amdgcn_wmma_f32_16x16x4_f32(false, a, false, b,
                                                   (short)0, acc0, false, false);
  }
  v8f acc = acc0 + acc1;

  // D tile == out[16 x 16] row-major: VGPR v -> rows v (lanes 0-15) / v+8.
  const int prow = base + (lo ? 0 : 8);
#pragma unroll
  for (int v = 0; v < 8; ++v)
    out[(size_t)(prow + v) * 16 + col] = acc[v];
}

// ======================================================================
extern "C" void kernel_launch(void* const* d_in, const int* in_sizes, int n_in,
                              void* d_out, int out_size, void* d_ws, size_t ws_size,
                              hipStream_t stream)
{
  (void)n_in; (void)out_size; (void)d_ws; (void)ws_size;
  const float* f10 = (const float*)d_in[0];
  const float* f11 = (const float*)d_in[1];
  const float* f12 = (const float*)d_in[2];
  const float* f13 = (const float*)d_in[3];
  const float* f20 = (const float*)d_in[4];
  const float* f21 = (const float*)d_in[5];
  const float* f22 = (const float*)d_in[6];
  const float* f23 = (const float*)d_in[7];
  float* out = (float*)d_out;

  const int NC = in_sizes[0];              // f1_l0 has N*C elements
  if (NC < 16) return;                     // reference shape is 2.56M; guard
  const int ntiles = (NC + 15) >> 4;       // 16 points per wave, last tile clamped
  dim3 grid((ntiles + 7) / 8);             // 8 waves (256 threads) per block
  cg_tp_wmma<<<grid, 256, 0, stream>>>(f10, f11, f12, f13,
                                       f20, f21, f22, f23, out, ntiles, NC);
}